// Pseudo_IMG_Scatter_22342419873900
// MI455X (gfx1250) — compile-verified
//
#include <hip/hip_runtime.h>
#include <cstdint>
#include <cstddef>

// ---------------------------------------------------------------------------
// Pseudo_IMG_Scatter for MI455X (gfx1250)
//   out[b, f, y*XS + x] += pillars[b, p, f] * (contains[b,p]==1)
// Memory-bound scatter: 256 MB output zero+scatter, 12 MB input stream.
// Zero-fill uses the CDNA5 Tensor Data Mover (TENSOR_STORE_FROM_LDS with
// descriptor iteration) so one DMA instruction per block streams 4 MB of
// zeros from a 32 KB LDS tile. Scatter uses wave32-coalesced loads and
// fire-and-forget global_atomic_add_f32.
// ---------------------------------------------------------------------------

#define TILE_BYTES  32768u                      // LDS zero tile (32 KB)
#define CHUNK_BYTES (4u * 1024u * 1024u)        // zero coverage per block (4 MB)
#define TDM_ITER    (CHUNK_BYTES / TILE_BYTES)  // 128 iterations
#define TILE_ELEMS  (TILE_BYTES / 8u)           // data_size = 8B units -> 4096

typedef unsigned int u32x4 __attribute__((ext_vector_type(4)));
typedef int          i32x4 __attribute__((ext_vector_type(4)));
typedef int          i32x8 __attribute__((ext_vector_type(8)));

#if defined(__has_builtin)
#if __has_builtin(__builtin_amdgcn_tensor_store_from_lds)
#define TDM_AVAILABLE 1
#endif
#endif

// --- Kernel 1: zero CHUNK_BYTES per block, TDM path with store fallback ----
__global__ void pp_zero_chunk_kernel(float* __restrict__ out) {
#if defined(__HIP_DEVICE_COMPILE__) && defined(TDM_AVAILABLE)
  extern __shared__ unsigned char lds_tile[];
  // Zero the 32 KB LDS tile cooperatively.
  float4 z; z.x = 0.f; z.y = 0.f; z.z = 0.f; z.w = 0.f;
  float4* l4 = reinterpret_cast<float4*>(lds_tile);
  for (unsigned i = threadIdx.x; i < TILE_BYTES / 16u; i += blockDim.x) l4[i] = z;
  __syncthreads();

  // One wave (wave-uniform branch) issues a single iterated TDM store:
  // LDS tile -> global, 128 iterations, global_addr += 32 KB each, LDS += 0.
  if (threadIdx.x < 32u) {
    uint64_t ga = (uint64_t)(uintptr_t)out + (uint64_t)blockIdx.x * CHUNK_BYTES;

    u32x4 g0 = {0u, 0u, 0u, 0u};
    g0[0] = 1u;                                   // count=1, is_restore=0
    g0[1] = 0u;                                   // lds_addr = 0 (dyn-LDS base)
    g0[2] = (unsigned)(ga & 0xFFFFFFFFull);       // global_addr[31:0]
    g0[3] = (unsigned)(ga >> 32) | 0x80000000u;   // addr[56:32] | type=2<<30

    i32x8 g1 = {0, 0, 0, 0, 0, 0, 0, 0};
    g1[0] = (3 << 16) | (1 << 19);                // data_size=8B, iterate_enable
    g1[1] = (int)(TILE_ELEMS << 16);              // tensor_dim0[15:0] @ [31:16]
    g1[2] = 0;                                    // tensor_dim0 hi / tensor_dim1 lo
    g1[3] = (int)(TILE_ELEMS << 16);              // tile_dim0 @ [31:16]
    g1[5] = (int)TILE_ELEMS;                      // tensor_dim0_stride[31:0]

    i32x4 g2 = {0, 0, 0, 0};
    g2[1] = 0;                                    // lds_addr_increment = 0
    g2[2] = (int)TILE_ELEMS;                      // global_addr_increment (elems)
    g2[3] = (int)((TDM_ITER - 1u) << 16);         // iterate_count @ [31:16]

    i32x4 g3 = {0, 0, 0, 0};

#if defined(__clang_major__) && (__clang_major__ >= 23)
    i32x8 g4 = {0, 0, 0, 0, 0, 0, 0, 0};
    __builtin_amdgcn_tensor_store_from_lds(g0, g1, g2, g3, g4, 0);
#else
    __builtin_amdgcn_tensor_store_from_lds(g0, g1, g2, g3, 0);
#endif
    __builtin_amdgcn_s_wait_tensorcnt(0);
  }
#else
  // Fallback: plain wide stores cover this block's 4 MB chunk.
  float4 z; z.x = 0.f; z.y = 0.f; z.z = 0.f; z.w = 0.f;
  float4* o4 = reinterpret_cast<float4*>(out) +
               (size_t)blockIdx.x * (CHUNK_BYTES / 16u);
  for (unsigned i = threadIdx.x; i < CHUNK_BYTES / 16u; i += blockDim.x) o4[i] = z;
#endif
}

// --- Kernel 2: zero any tail not covered by whole chunks -------------------
__global__ void pp_zero_tail_kernel(float* __restrict__ out, size_t n) {
  size_t stride = (size_t)gridDim.x * blockDim.x;
  for (size_t i = (size_t)blockIdx.x * blockDim.x + threadIdx.x; i < n; i += stride)
    out[i] = 0.0f;
}

// --- Kernel 3: masked scatter-add ------------------------------------------
// thread = (pillar_global, feature); consecutive threads = consecutive
// features -> coalesced b32 loads, wave-uniform contains-mask branch (NF>=32).
__global__ void pp_scatter_kernel(const float* __restrict__ pillars,
                                  const int*   __restrict__ coord,
                                  const int*   __restrict__ contains,
                                  const int*   __restrict__ xs_p,
                                  const int*   __restrict__ ys_p,
                                  float*       __restrict__ out,
                                  int BP, int NF, int out_elems) {
  int pf = blockIdx.x * blockDim.x + threadIdx.x;
  if (pf >= BP * NF) return;
  int pg = pf / NF;                 // global pillar index in [0, B*P)
  int f  = pf - pg * NF;            // feature index
  if (contains[pg] != 1) return;    // wave-uniform: whole wave skips

  int xs    = xs_p[0];
  int ys    = ys_p[0];
  int plane = xs * ys;
  int B     = out_elems / (NF * plane);
  int P     = BP / B;
  int b     = pg / P;

  int y = coord[pg * 3 + 1];
  int x = coord[pg * 3 + 2];

  float v = __builtin_nontemporal_load(pillars + (size_t)pg * NF + f);

  size_t off = (size_t)(b * NF + f) * (size_t)plane + (size_t)y * xs + (size_t)x;
  unsafeAtomicAdd(out + off, v);    // global_atomic_add_f32, no return
}

// ---------------------------------------------------------------------------
extern "C" void kernel_launch(void* const* d_in, const int* in_sizes, int n_in,
                              void* d_out, int out_size, void* d_ws, size_t ws_size,
                              hipStream_t stream) {
  const float* pillars  = (const float*)d_in[0];
  const int*   coord    = (const int*)  d_in[1];
  const int*   contains = (const int*)  d_in[2];
  const int*   xs_p     = (const int*)  d_in[3];
  const int*   ys_p     = (const int*)  d_in[4];
  float*       out      = (float*)      d_out;

  const int BP = in_sizes[2];            // B * P
  const int NF = in_sizes[0] / BP;       // features per pillar

  // Pass 1: zero the output (TDM DMA per 4 MB chunk + tail).
  size_t out_bytes = (size_t)out_size * sizeof(float);
  size_t nchunks   = out_bytes / CHUNK_BYTES;
  if (nchunks) {
    pp_zero_chunk_kernel<<<dim3((unsigned)nchunks), 256, TILE_BYTES, stream>>>(out);
  }
  size_t done = nchunks * (CHUNK_BYTES / sizeof(float));
  size_t rem  = (size_t)out_size - done;
  if (rem) {
    unsigned tb = (unsigned)((rem + 256 * 256 - 1) / (256 * 256));
    if (tb > 2048u) tb = 2048u;
    if (tb == 0u) tb = 1u;
    pp_zero_tail_kernel<<<dim3(tb), 256, 0, stream>>>(out + done, rem);
  }

  // Pass 2: masked scatter-add.
  long long total  = (long long)BP * NF;
  unsigned  blocks = (unsigned)((total + 255) / 256);
  pp_scatter_kernel<<<dim3(blocks), 256, 0, stream>>>(
      pillars, coord, contains, xs_p, ys_p, out, BP, NF, out_size);
}